// StreamingDurationProjector_35665408426348
// MI455X (gfx1250) — compile-verified
//
#include <hip/hip_runtime.h>

// ---------------------------------------------------------------------------
// StreamingDurationProjector for MI455X (gfx1250, wave32)
//
// Kernel 1: masked row-sum (prefix_total per row) via V_WMMA_F32_16X16X4_F32.
//           One wave handles 16 rows; B = all-ones, A = masked src chunk.
// Kernel 2: per-row sequential scan; one lane per row, branch-free body,
//           early exit at committed_len, global_prefetch_b8 ahead of loads.
// ---------------------------------------------------------------------------

typedef __attribute__((ext_vector_type(2))) float v2f;
typedef __attribute__((ext_vector_type(8))) float v8f;

#define ROWS_PER_WAVE 16

// ---------------------------------------------------------------------------
// Kernel 1: prefix_total[b] = sum_{i < cl[b]} src[b][i]   (fp32 WMMA rowsum)
// ---------------------------------------------------------------------------
__global__ __launch_bounds__(32)
void sdp_prefix_wmma(const float* __restrict__ src,
                     const int*   __restrict__ cl,
                     float*       __restrict__ prefix_total,
                     int N)
{
    const int lane = threadIdx.x;          // 0..31, wave32
    const int half = lane >> 4;            // 0: K=0,1  1: K=2,3  (ISA A-layout 16x4 f32)
    const int m    = lane & 15;            // row-in-tile
    const int rowbase = blockIdx.x * ROWS_PER_WAVE;
    const int row  = rowbase + m;
    const int koff = half * 2;

    const int   cli = cl[row];
    const float* rp = src + (size_t)row * N;

    v8f acc = {0.f, 0.f, 0.f, 0.f, 0.f, 0.f, 0.f, 0.f};
    v2f bones;                             // B = 4x16 all-ones: layout-independent
    bones[0] = 1.0f;
    bones[1] = 1.0f;

    for (int k = 0; k < N; k += 4) {
        const int k0 = k + koff;
        float s0 = rp[k0];
        float s1 = rp[k0 + 1];
        v2f a;
        a[0] = (k0     < cli) ? s0 : 0.0f;   // contiguous-prefix commit mask
        a[1] = (k0 + 1 < cli) ? s1 : 0.0f;
        // D = A x B + C : emits v_wmma_f32_16x16x4_f32
        acc = __builtin_amdgcn_wmma_f32_16x16x4_f32(
            /*neg_a=*/false, a, /*neg_b=*/false, bones,
            /*c_mod=*/(short)0, acc, /*reuse_a=*/false, /*reuse_b=*/false);
    }

    // 16x16 f32 D layout: VGPR r -> lanes 0-15 hold M=r, lanes 16-31 hold M=8+r.
    // Every column of row m equals rowsum(m); read column N=0 (lanes 0 and 16).
    if (lane == 0) {
        prefix_total[rowbase + 0] = acc[0];
        prefix_total[rowbase + 1] = acc[1];
        prefix_total[rowbase + 2] = acc[2];
        prefix_total[rowbase + 3] = acc[3];
        prefix_total[rowbase + 4] = acc[4];
        prefix_total[rowbase + 5] = acc[5];
        prefix_total[rowbase + 6] = acc[6];
        prefix_total[rowbase + 7] = acc[7];
    } else if (lane == 16) {
        prefix_total[rowbase +  8] = acc[0];
        prefix_total[rowbase +  9] = acc[1];
        prefix_total[rowbase + 10] = acc[2];
        prefix_total[rowbase + 11] = acc[3];
        prefix_total[rowbase + 12] = acc[4];
        prefix_total[rowbase + 13] = acc[5];
        prefix_total[rowbase + 14] = acc[6];
        prefix_total[rowbase + 15] = acc[7];
    }
}

// ---------------------------------------------------------------------------
// Kernel 2: per-row scan. One lane per row.
// ---------------------------------------------------------------------------
__global__ __launch_bounds__(32)
void sdp_scan(const float* __restrict__ ex,   // unit_duration_exec
              const float* __restrict__ sr,   // source_duration_obs
              const int*   __restrict__ sp,   // speech_mask
              const int*   __restrict__ bd,   // boundary_cue
              const int*   __restrict__ cl,   // committed_len
              const float* __restrict__ r0,   // residual_prev
              const float* __restrict__ o0,   // prefix_offset_prev
              const float* __restrict__ ptotal,
              float*       __restrict__ proj,
              float*       __restrict__ cf,
              float*       __restrict__ of,
              int Brows, int N)
{
    const int b = blockIdx.x * 32 + threadIdx.x;
    if (b >= Brows) return;

    const int c = cl[b];

    // _resolve_prefix_budget (dynamic path when committed_len > 0)
    float dyn = rintf(ptotal[b] * 0.25f);          // round half-to-even, as jnp.round
    dyn = fmaxf(dyn, 8.0f);                        // MIN_BUDGET
    dyn = fminf(dyn, 96.0f);                       // MAX_BUDGET
    dyn = fmaxf(dyn, 0.0f);
    const float budget = (c > 0) ? dyn : 24.0f;    // BUDGET_POS == BUDGET_NEG
    const float bpos = budget;
    const float bneg = budget;

    float carry = r0[b];
    float off   = o0[b];

    const size_t base = (size_t)b * N;
    const float* ep  = ex + base;
    const float* srp = sr + base;
    const int*   spp = sp + base;
    const int*   bdp = bd + base;
    float*       pp  = proj + base;

    // clamp prefetch targets to stay inside the B*N allocations
    const size_t last = (size_t)Brows * N - 1;
    const float* ep_max  = ex + last;
    const float* srp_max = sr + last;

    // Active region is a contiguous prefix [0, c): beyond it proj=0 and the
    // (carry, off) state is never updated, so we exit the real loop at c.
    for (int i = 0; i < c; ++i) {
        // independent of the carry chain -> free issue slots; emits global_prefetch_b8
        const float* pe = ep  + i + 256; pe = (pe < ep_max)  ? pe : ep_max;
        const float* ps = srp + i + 256; ps = (ps < srp_max) ? ps : srp_max;
        __builtin_prefetch(pe, 0, 0);
        __builtin_prefetch(ps, 0, 0);

        const float e  = ep[i];
        const float s  = srp[i];
        const int   spv = spp[i];
        const int   bdv = bdp[i];

        const float src_cnt = fmaxf(0.0f, rintf(s));
        const float anchor  = fmaxf(1.0f, src_cnt);
        const float total   = fmaxf(0.0f, e + carry);
        float frames        = fmaxf(1.0f, floorf(total + 0.5f));
        const float lower   = fmaxf(1.0f, ceilf(anchor - (bneg + off)));
        const float upper   = fmaxf(lower, floorf(anchor + (bpos - off)));
        frames = fminf(fmaxf(frames, lower), upper);          // jnp.clip

        float n_off   = off + frames - anchor;
        float n_carry = total - frames;
        if (bdv > 0) {                                        // boundary decay
            n_carry *= 0.25f;
            n_off = fminf(fmaxf(rintf(n_off * 0.25f), -bneg), bpos);
        }
        const bool upd = (spv > 0);                           // active==true here
        carry = upd ? n_carry : carry;
        off   = upd ? n_off   : off;
        pp[i] = (spv > 0) ? frames : src_cnt;
    }

    // inactive tail: projection is zero, state untouched
    for (int i = c; i < N; ++i) pp[i] = 0.0f;

    cf[b] = carry;
    of[b] = off;
}

// ---------------------------------------------------------------------------
// Launch
// ---------------------------------------------------------------------------
extern "C" void kernel_launch(void* const* d_in, const int* in_sizes, int n_in,
                              void* d_out, int out_size, void* d_ws, size_t ws_size,
                              hipStream_t stream) {
    const float* ex = (const float*)d_in[0];   // unit_duration_exec (B,N) f32
    const float* sr = (const float*)d_in[1];   // source_duration_obs (B,N) f32
    const int*   sp = (const int*)  d_in[2];   // speech_mask (B,N) i32
    const int*   bd = (const int*)  d_in[3];   // boundary_cue (B,N) i32
    const int*   cl = (const int*)  d_in[4];   // committed_len (B,) i32
    const float* r0 = (const float*)d_in[5];   // residual_prev (B,1) f32
    const float* o0 = (const float*)d_in[6];   // prefix_offset_prev (B,1) f32

    const int Brows = in_sizes[4];
    const int N     = in_sizes[0] / Brows;

    float* proj = (float*)d_out;               // (B,N)
    float* cf   = proj + (size_t)Brows * N;    // (B,1)
    float* of   = cf + Brows;                  // (B,1)

    float* ptotal = (float*)d_ws;              // B floats of scratch

    // Kernel 1: WMMA rowsums. One wave (32 threads) per 16 rows.
    sdp_prefix_wmma<<<dim3(Brows / ROWS_PER_WAVE), dim3(32), 0, stream>>>(
        sr, cl, ptotal, N);

    // Kernel 2: per-row scan. Single-wave workgroups to spread across WGPs.
    sdp_scan<<<dim3((Brows + 31) / 32), dim3(32), 0, stream>>>(
        ex, sr, sp, bd, cl, r0, o0, ptotal, proj, cf, of, Brows, N);
}